// HausdorffDistanceLoss_59296318488926
// MI455X (gfx1250) — compile-verified
//
#include <hip/hip_runtime.h>
#include <hip/hip_bf16.h>
#include <math.h>

// Hausdorff distance, B=8, N=M=4096, D=3.
// d2(i,j) = |p_i|^2 + (-2 p.t + |t_j|^2); bilinear part via V_WMMA_F32_16X16X4_F32:
//   A row (K=4): [p_x, p_y, p_z, 1]
//   B col (K=4): [-2 t_x, -2 t_y, -2 t_z, |t|^2]
// Row bias |p_i|^2 added after the min-over-j reduction (constant per row).

typedef __attribute__((ext_vector_type(2))) float v2f;
typedef __attribute__((ext_vector_type(8))) float v8f;

#define NPTS        4096
#define NTILES      (NPTS / 16)      // 256 column tiles
#define CHUNK_TILES 128              // 32 KB LDS per chunk
#define WAVES_PER_WG 8
#define WG_THREADS   256
#define NBATCH       8

__global__ void hd_init_kernel(float* ws) {
    if (threadIdx.x < 2 * NBATCH) ws[threadIdx.x] = 0.0f;
}

__global__ __launch_bounds__(WG_THREADS)
void hd_directed_kernel(const float* __restrict__ pred,
                        const float* __restrict__ targ,
                        float* __restrict__ ws) {
    const int b   = blockIdx.y;
    const int dir = blockIdx.z;
    const float* Aset = (dir == 0 ? pred : targ) + (size_t)b * NPTS * 3;
    const float* Bset = (dir == 0 ? targ : pred) + (size_t)b * NPTS * 3;

    // LDS staging of B operands in WMMA lane layout:
    // entry (tile jt, lane l): .x = v0 operand, .y = v1 operand
    //   lanes 0-15  : v0 = K0 row = -2*t_x, v1 = K1 row = -2*t_y   (N = l)
    //   lanes 16-31 : v0 = K2 row = -2*t_z, v1 = K3 row = |t|^2    (N = l-16)
    __shared__ float2 sB[CHUNK_TILES * 32];

    const int tid  = threadIdx.x;
    const int lane = tid & 31;
    const int wave = tid >> 5;
    const int rowBase = (blockIdx.x * WAVES_PER_WG + wave) * 16;
    const int myRow   = rowBase + (lane & 15);

    // A operand: 32-bit A 16x4 layout — v0: K0 (lanes 0-15) / K2 (lanes 16-31),
    //                                   v1: K1 (lanes 0-15) / K3 (lanes 16-31)
    const float px = Aset[myRow * 3 + 0];
    const float py = Aset[myRow * 3 + 1];
    const float pz = Aset[myRow * 3 + 2];
    v2f a;
    a[0] = (lane < 16) ? px : pz;
    a[1] = (lane < 16) ? py : 1.0f;
    const float mynorm = px * px + py * py + pz * pz;   // |p_{rowBase + lane%16}|^2

    float acc[8];
#pragma unroll
    for (int r = 0; r < 8; ++r) acc[r] = 3.0e38f;

    for (int chunk = 0; chunk < NTILES; chunk += CHUNK_TILES) {
        __syncthreads();   // previous chunk's consumers done
        for (int e = tid; e < CHUNK_TILES * 32; e += WG_THREADS) {
            const int jt = e >> 5;
            const int el = e & 31;
            const int j  = (chunk + jt) * 16 + (el & 15);
            const float tx = Bset[j * 3 + 0];
            const float ty = Bset[j * 3 + 1];
            const float tz = Bset[j * 3 + 2];
            float2 v;
            if (el < 16) { v.x = -2.0f * tx; v.y = -2.0f * ty; }
            else         { v.x = -2.0f * tz; v.y = tx * tx + ty * ty + tz * tz; }
            sB[e] = v;
        }
        __syncthreads();

#pragma unroll 4
        for (int jt = 0; jt < CHUNK_TILES; ++jt) {
            const float2 bv = sB[jt * 32 + lane];   // one ds_load_b64 per tile
            v2f bm; bm[0] = bv.x; bm[1] = bv.y;
            v8f c = {};
            v8f d = __builtin_amdgcn_wmma_f32_16x16x4_f32(
                false, a, false, bm, (short)0, c, false, false);
#pragma unroll
            for (int r = 0; r < 8; ++r) acc[r] = fminf(acc[r], d[r]);
        }
    }

    // acc[r] lane l holds min over {j : j%16 == l%16} for row M = r + 8*(l/16).
    // Reduce across the 16 lanes of each half -> true row min.
#pragma unroll
    for (int r = 0; r < 8; ++r) {
#pragma unroll
        for (int off = 1; off < 16; off <<= 1)
            acc[r] = fminf(acc[r], __shfl_xor(acc[r], off, 32));
    }

    // Add |p_i|^2 bias (row i = r + 8*(lane/16); its norm lives at lane (r + 8*(lane/16))).
    const int hsel = (lane >> 4) << 3;    // 0 or 8
    float m = -3.0e38f;
#pragma unroll
    for (int r = 0; r < 8; ++r) {
        const float bias = __shfl(mynorm, r + hsel, 32);
        m = fmaxf(m, acc[r] + bias);
    }
    m = fmaxf(m, __shfl_xor(m, 16, 32));  // combine the two row halves
    m = fmaxf(m, 0.0f);                   // clamp: keeps uint ordering valid

    if (lane == 0)
        atomicMax((unsigned int*)&ws[b * 2 + dir], __float_as_uint(m));
}

__global__ void hd_final_kernel(const float* __restrict__ ws,
                                float* __restrict__ out) {
    if (threadIdx.x == 0 && blockIdx.x == 0) {
        float s = 0.0f;
        for (int b = 0; b < NBATCH; ++b) {
            const float f2 = ws[b * 2 + 0];
            const float w2 = ws[b * 2 + 1];
            s += sqrtf(fmaxf(f2, w2));    // max(sqrt(a),sqrt(b)) = sqrt(max(a,b))
        }
        out[0] = s / (float)NBATCH;
    }
}

extern "C" void kernel_launch(void* const* d_in, const int* in_sizes, int n_in,
                              void* d_out, int out_size, void* d_ws, size_t ws_size,
                              hipStream_t stream) {
    (void)in_sizes; (void)n_in; (void)out_size; (void)ws_size;
    const float* pred = (const float*)d_in[0];
    const float* targ = (const float*)d_in[1];
    float* ws  = (float*)d_ws;
    float* out = (float*)d_out;

    hipLaunchKernelGGL(hd_init_kernel, dim3(1), dim3(32), 0, stream, ws);

    dim3 grid(NPTS / (16 * WAVES_PER_WG), NBATCH, 2);   // 32 x 8 x 2 workgroups
    hipLaunchKernelGGL(hd_directed_kernel, grid, dim3(WG_THREADS), 0, stream,
                       pred, targ, ws);

    hipLaunchKernelGGL(hd_final_kernel, dim3(1), dim3(32), 0, stream, ws, out);
}